// DGCNN_18614388261213
// MI455X (gfx1250) — compile-verified
//
#include <hip/hip_runtime.h>
#include <math.h>

typedef __attribute__((ext_vector_type(2))) float v2f;
typedef __attribute__((ext_vector_type(4))) float v4f;
typedef __attribute__((ext_vector_type(8))) float v8f;

#define NEG_INF -3.402823466e38f

// Branchless top-3 (value, index) update; strict > keeps earliest index on ties
// (matches lax.top_k stability when scanned in increasing index order).
__device__ __forceinline__ void top3_update(float val, int idx,
                                            float& b0, float& b1, float& b2,
                                            int& i0, int& i1, int& i2) {
  const bool gt0 = val > b0;
  const bool gt1 = val > b1;
  const bool gt2 = val > b2;
  const float nb2 = gt1 ? b1 : (gt2 ? val : b2);
  const int   ni2 = gt1 ? i1 : (gt2 ? idx : i2);
  const float nb1 = gt0 ? b0 : (gt1 ? val : b1);
  const int   ni1 = gt0 ? i0 : (gt1 ? idx : i1);
  b0 = gt0 ? val : b0;
  i0 = gt0 ? idx : i0;
  b1 = nb1; i1 = ni1;
  b2 = nb2; i2 = ni2;
}

// ---------------------------------------------------------------------------
// Kernel 1: KNN (top-3) over 4096 points per batch using WMMA f32 16x16x4 for
// the gram tiles. score(m) = 2*dot(x_n,x_m) - ||x_m||^2  (row-constant
// dropped; ranking identical to reference dist).
// Block = 256 (8 waves); wave owns 16 rows; lo lanes scan cols 0-7 of a row,
// hi lanes cols 8-15; partial top-3 lists merged once at the end.
// Grid = (32, B).
// ---------------------------------------------------------------------------
#define TPITCH 20
__global__ __launch_bounds__(256) void dgcnn_knn_kernel(
    const float* __restrict__ xg, int* __restrict__ knn_idx) {
  __shared__ float Xs[3 * 4096];                 // x_resh[b][c][n]
  __shared__ alignas(16) float tile[8][16 * TPITCH];

  const int b  = blockIdx.y;
  const int n0 = blockIdx.x * 128;
  const int t  = threadIdx.x;

  const float* xb = xg + (size_t)b * 12288;
  for (int j = t; j < 12288; j += 256) Xs[j] = xb[j];
  __syncthreads();

  const int w    = t >> 5;
  const int lane = t & 31;
  const int ml   = lane & 15;
  const bool lo  = lane < 16;

  // A tile (loop invariant). 16x4 f32 A layout:
  // lanes 0-15 -> M=lane, v0=K0, v1=K1 ; lanes 16-31 -> M=lane-16, v0=K2, v1=K3
  const int rowA = n0 + w * 16 + ml;
  const float ar0 = Xs[rowA];
  const float ar1 = Xs[4096 + rowA];
  const float ar2 = Xs[8192 + rowA];
  v2f a;
  a.x = lo ? ar0 : ar2;
  a.y = lo ? ar1 : 0.0f;

  float b0 = NEG_INF, b1 = NEG_INF, b2 = NEG_INF;
  int   i0 = 0,       i1 = 0,       i2 = 0;

  float* myTile = tile[w];
  const int colHalf = lo ? 0 : 8;                  // which 8 columns this lane scans
  const v4f* rowPtr = (const v4f*)(myTile + ml * TPITCH + colHalf);

  for (int m0 = 0; m0 < 4096; m0 += 16) {
    const int colB = m0 + ml;
    const float c0 = Xs[colB];
    const float c1 = Xs[4096 + colB];
    const float c2 = Xs[8192 + colB];
    v2f bv;
    bv.x = lo ? c0 : c2;
    bv.y = lo ? c1 : 0.0f;
    const float smc = -(c0 * c0 + c1 * c1 + c2 * c2);   // -||x_m||^2

    v8f cz = {0.f, 0.f, 0.f, 0.f, 0.f, 0.f, 0.f, 0.f};
    v8f cc = __builtin_amdgcn_wmma_f32_16x16x4_f32(
        false, a, false, bv, (short)0, cz, false, false);

    // C layout: lane<16 -> col=lane, rows 0..7 ; lane>=16 -> col=lane-16, rows 8..15
    const int rbase = lo ? 0 : 8;
#pragma unroll
    for (int v = 0; v < 8; ++v) {
      myTile[(rbase + v) * TPITCH + ml] = 2.0f * cc[v] + smc;
    }
    asm volatile("s_wait_dscnt 0" ::: "memory");

    // Scan 8 columns of row (lane&15) as two b128 LDS loads, branchless top-3.
    const v4f q0 = rowPtr[0];
    const v4f q1 = rowPtr[1];
    const int bi = m0 + colHalf;
    top3_update(q0.x, bi + 0, b0, b1, b2, i0, i1, i2);
    top3_update(q0.y, bi + 1, b0, b1, b2, i0, i1, i2);
    top3_update(q0.z, bi + 2, b0, b1, b2, i0, i1, i2);
    top3_update(q0.w, bi + 3, b0, b1, b2, i0, i1, i2);
    top3_update(q1.x, bi + 4, b0, b1, b2, i0, i1, i2);
    top3_update(q1.y, bi + 5, b0, b1, b2, i0, i1, i2);
    top3_update(q1.z, bi + 6, b0, b1, b2, i0, i1, i2);
    top3_update(q1.w, bi + 7, b0, b1, b2, i0, i1, i2);
  }

  // Merge lo-half (cols ≡ 0..7 mod 16) and hi-half (cols ≡ 8..15) top-3 lists.
  const float ob0 = __shfl_xor(b0, 16, 32);
  const float ob1 = __shfl_xor(b1, 16, 32);
  const float ob2 = __shfl_xor(b2, 16, 32);
  const int   oi0 = __shfl_xor(i0, 16, 32);
  const int   oi1 = __shfl_xor(i1, 16, 32);
  const int   oi2 = __shfl_xor(i2, 16, 32);

  if (lo) {
    const float av[3] = {b0, b1, b2};
    const int   ai[3] = {i0, i1, i2};
    const float cv[3] = {ob0, ob1, ob2};
    const int   ci[3] = {oi0, oi1, oi2};
    int pa = 0, pb = 0;
    int ri[3];
#pragma unroll
    for (int k = 0; k < 3; ++k) {
      const float ax  = (pa == 0) ? av[0] : ((pa == 1) ? av[1] : av[2]);
      const int   aix = (pa == 0) ? ai[0] : ((pa == 1) ? ai[1] : ai[2]);
      const float bx  = (pb == 0) ? cv[0] : ((pb == 1) ? cv[1] : cv[2]);
      const int   bix = (pb == 0) ? ci[0] : ((pb == 1) ? ci[1] : ci[2]);
      const bool takeA = (ax > bx) || ((ax == bx) && (aix < bix));
      ri[k] = takeA ? aix : bix;
      pa += takeA ? 1 : 0;
      pb += takeA ? 0 : 1;
    }
    const int row = n0 + w * 16 + ml;
    const size_t base = ((size_t)b * 4096 + row) * 3;
    knn_idx[base + 0] = ri[0];
    knn_idx[base + 1] = ri[1];
    knn_idx[base + 2] = ri[2];
  }
}

// ---------------------------------------------------------------------------
// Kernel 2: gather features, 1x1 conv (6->64), BN + leaky_relu(0.2),
// max over N -> x1[b][64][3].  Grid = (B, 3), block = 256.
// f[b][c][kk][n]: g = c*12288 + kk*4096 + n ; ns=g/18, ks=(g%18)/6, cs=g%6
// F = cs<3 ? x[b][idx[ns][ks]][cs] - x[b][ns][cs] : x[b][ns][cs-3]
// ---------------------------------------------------------------------------
__global__ __launch_bounds__(256) void dgcnn_conv_max_kernel(
    const float* __restrict__ xg, const int* __restrict__ knn_idx,
    const float* __restrict__ w1,
    const float* __restrict__ bng, const float* __restrict__ bnb,
    const float* __restrict__ bnm, const float* __restrict__ bnv,
    float* __restrict__ x1_ws) {
  __shared__ float w1s[64 * 6];
  __shared__ float scale[64], shift[64];
  __shared__ float red[8 * 64];

  const int b  = blockIdx.x;
  const int kk = blockIdx.y;
  const int t  = threadIdx.x;

  if (t < 384) w1s[t] = w1[t];
  if (t < 64) {
    const float sc = bng[t] * rsqrtf(bnv[t] + 1e-5f);
    scale[t] = sc;
    shift[t] = bnb[t] - bnm[t] * sc;
  }
  __syncthreads();

  const float* xb = xg + (size_t)b * 12288;
  const int*   ib = knn_idx + (size_t)b * 4096 * 3;

  float acc[64];
#pragma unroll
  for (int o = 0; o < 64; ++o) acc[o] = NEG_INF;

  for (int n = t; n < 4096; n += 256) {
    float F[6];
#pragma unroll
    for (int c = 0; c < 6; ++c) {
      const int g   = c * 12288 + kk * 4096 + n;
      const int ns  = g / 18;
      const int rem = g - ns * 18;
      const int ks  = rem / 6;
      const int cs  = rem - ks * 6;
      if (cs < 3) {
        const int nb = ib[ns * 3 + ks];
        F[c] = xb[nb * 3 + cs] - xb[ns * 3 + cs];
      } else {
        F[c] = xb[ns * 3 + cs - 3];
      }
    }
#pragma unroll
    for (int o = 0; o < 64; ++o) {
      float h = w1s[o * 6 + 0] * F[0] + w1s[o * 6 + 1] * F[1] +
                w1s[o * 6 + 2] * F[2] + w1s[o * 6 + 3] * F[3] +
                w1s[o * 6 + 4] * F[4] + w1s[o * 6 + 5] * F[5];
      h = h * scale[o] + shift[o];
      h = h > 0.0f ? h : 0.2f * h;
      acc[o] = fmaxf(acc[o], h);
    }
  }

  // wave32 max reduction
#pragma unroll
  for (int o = 0; o < 64; ++o) {
    float v = acc[o];
    for (int off = 16; off >= 1; off >>= 1)
      v = fmaxf(v, __shfl_xor(v, off, 32));
    acc[o] = v;
  }
  const int w = t >> 5, lane = t & 31;
  if (lane == 0) {
    for (int o = 0; o < 64; ++o) red[w * 64 + o] = acc[o];
  }
  __syncthreads();
  if (t < 64) {
    float v = red[t];
    for (int w2 = 1; w2 < 8; ++w2) v = fmaxf(v, red[w2 * 64 + t]);
    x1_ws[((size_t)b * 64 + t) * 3 + kk] = v;  // x1[b][o][kk]
  }
}

// ---------------------------------------------------------------------------
// Kernel 3: tail — x2/x3/x4 (tiny knn+feature+max), concat, MLP, softmax.
// One block per batch.
// ---------------------------------------------------------------------------
__global__ __launch_bounds__(256) void dgcnn_tail_kernel(
    const float* __restrict__ x1_ws,
    const float* __restrict__ wA, const float* __restrict__ bA,
    const float* __restrict__ wB, const float* __restrict__ bB,
    const float* __restrict__ wC, const float* __restrict__ bC,
    const float* __restrict__ bnAg, const float* __restrict__ bnAb,
    const float* __restrict__ bnAm, const float* __restrict__ bnAv,
    const float* __restrict__ bnBg, const float* __restrict__ bnBb,
    const float* __restrict__ bnBm, const float* __restrict__ bnBv,
    float* __restrict__ out) {
  __shared__ float x1s[192];
  __shared__ int   idxA[64 * 3];
  __shared__ float x2s[18], x3s[18], x4s[18];
  __shared__ int   idxB[6 * 3];
  __shared__ float p[82];
  __shared__ float hA[256];
  __shared__ float hB[128];
  __shared__ float lg[3];

  const int b = blockIdx.x;
  const int t = threadIdx.x;

  if (t < 192) x1s[t] = x1_ws[(size_t)b * 192 + t];
  __syncthreads();

  // knn over 64 "points": xr[c][j] = x1s[c*64+j]
  if (t < 64) {
    const float a0 = x1s[t], a1 = x1s[64 + t], a2 = x1s[128 + t];
    float b0 = NEG_INF, b1 = NEG_INF, b2 = NEG_INF;
    int i0 = 0, i1 = 0, i2 = 0;
    for (int m = 0; m < 64; ++m) {
      const float c0 = x1s[m], c1 = x1s[64 + m], c2 = x1s[128 + m];
      const float val = 2.f * (a0 * c0 + a1 * c1 + a2 * c2) -
                        (c0 * c0 + c1 * c1 + c2 * c2);
      top3_update(val, m, b0, b1, b2, i0, i1, i2);
    }
    idxA[t * 3] = i0; idxA[t * 3 + 1] = i1; idxA[t * 3 + 2] = i2;
  }
  __syncthreads();

  // x2[o][kk] = max_n F ; g = o*192 + kk*64 + n
  if (t < 18) {
    const int o = t / 3, kk = t - o * 3;
    float mx = NEG_INF;
    for (int n = 0; n < 64; ++n) {
      const int g = o * 192 + kk * 64 + n;
      const int ns = g / 18, rem = g - ns * 18;
      const int ks = rem / 6, cs = rem - ks * 6;
      const float F = (cs < 3)
          ? x1s[idxA[ns * 3 + ks] * 3 + cs] - x1s[ns * 3 + cs]
          : x1s[ns * 3 + cs - 3];
      mx = fmaxf(mx, F);
    }
    x2s[t] = mx;
  }
  __syncthreads();

  // small knn over 6 points
  auto knn6 = [&](const float* xs) {
    if (t < 6) {
      const float a0 = xs[t], a1 = xs[6 + t], a2 = xs[12 + t];
      float b0 = NEG_INF, b1 = NEG_INF, b2 = NEG_INF;
      int i0 = 0, i1 = 0, i2 = 0;
      for (int m = 0; m < 6; ++m) {
        const float c0 = xs[m], c1 = xs[6 + m], c2 = xs[12 + m];
        const float val = 2.f * (a0 * c0 + a1 * c1 + a2 * c2) -
                          (c0 * c0 + c1 * c1 + c2 * c2);
        top3_update(val, m, b0, b1, b2, i0, i1, i2);
      }
      idxB[t * 3] = i0; idxB[t * 3 + 1] = i1; idxB[t * 3 + 2] = i2;
    }
  };
  auto feat6 = [&](const float* xs, float* xd) {
    if (t < 18) {
      const int o = t / 3, kk = t - o * 3;
      float mx = NEG_INF;
      for (int n = 0; n < 6; ++n) {
        const int g = o * 18 + kk * 6 + n;
        const int ns = g / 18, rem = g - ns * 18;
        const int ks = rem / 6, cs = rem - ks * 6;
        const float F = (cs < 3)
            ? xs[idxB[ns * 3 + ks] * 3 + cs] - xs[ns * 3 + cs]
            : xs[ns * 3 + cs - 3];
        mx = fmaxf(mx, F);
      }
      xd[t] = mx;
    }
  };

  knn6(x2s); __syncthreads();
  feat6(x2s, x3s); __syncthreads();
  knn6(x3s); __syncthreads();
  feat6(x3s, x4s); __syncthreads();

  // p[82] = max over kk of concat(x1, x2, x3, x4)
  if (t < 82) {
    float v;
    if (t < 64)      v = fmaxf(fmaxf(x1s[t * 3], x1s[t * 3 + 1]), x1s[t * 3 + 2]);
    else if (t < 70) { const int o = t - 64; v = fmaxf(fmaxf(x2s[o * 3], x2s[o * 3 + 1]), x2s[o * 3 + 2]); }
    else if (t < 76) { const int o = t - 70; v = fmaxf(fmaxf(x3s[o * 3], x3s[o * 3 + 1]), x3s[o * 3 + 2]); }
    else             { const int o = t - 76; v = fmaxf(fmaxf(x4s[o * 3], x4s[o * 3 + 1]), x4s[o * 3 + 2]); }
    p[t] = v;
  }
  __syncthreads();

  // MLP A: 82 -> 256, BN, lrelu   (all 256 threads)
  {
    float acc = bA[t];
    for (int c = 0; c < 82; ++c) acc += p[c] * wA[t * 82 + c];
    const float sc = bnAg[t] * rsqrtf(bnAv[t] + 1e-5f);
    float hv = (acc - bnAm[t]) * sc + bnAb[t];
    hA[t] = hv > 0.f ? hv : 0.2f * hv;
  }
  __syncthreads();

  // MLP B: 256 -> 128, BN, lrelu
  if (t < 128) {
    float acc = bB[t];
    for (int c = 0; c < 256; ++c) acc += hA[c] * wB[t * 256 + c];
    const float sc = bnBg[t] * rsqrtf(bnBv[t] + 1e-5f);
    float hv = (acc - bnBm[t]) * sc + bnBb[t];
    hB[t] = hv > 0.f ? hv : 0.2f * hv;
  }
  __syncthreads();

  if (t < 3) {
    float acc = bC[t];
    for (int c = 0; c < 128; ++c) acc += hB[c] * wC[t * 128 + c];
    lg[t] = acc;
  }
  __syncthreads();

  if (t == 0) {
    const float mx = fmaxf(lg[0], fmaxf(lg[1], lg[2]));
    const float e0 = expf(lg[0] - mx), e1 = expf(lg[1] - mx), e2 = expf(lg[2] - mx);
    const float s = e0 + e1 + e2;
    out[b * 3 + 0] = e0 / s;
    out[b * 3 + 1] = e1 / s;
    out[b * 3 + 2] = e2 / s;
  }
}

extern "C" void kernel_launch(void* const* d_in, const int* in_sizes, int n_in,
                              void* d_out, int out_size, void* d_ws, size_t ws_size,
                              hipStream_t stream) {
  const float* xg  = (const float*)d_in[0];
  const float* w1  = (const float*)d_in[1];
  const float* wA  = (const float*)d_in[2];
  const float* bA  = (const float*)d_in[3];
  const float* wB  = (const float*)d_in[4];
  const float* bB  = (const float*)d_in[5];
  const float* wC  = (const float*)d_in[6];
  const float* bC  = (const float*)d_in[7];
  const float* bn1g = (const float*)d_in[8];
  const float* bn1b = (const float*)d_in[9];
  const float* bn1m = (const float*)d_in[10];
  const float* bn1v = (const float*)d_in[11];
  const float* bnAg = (const float*)d_in[12];
  const float* bnAb = (const float*)d_in[13];
  const float* bnAm = (const float*)d_in[14];
  const float* bnAv = (const float*)d_in[15];
  const float* bnBg = (const float*)d_in[16];
  const float* bnBb = (const float*)d_in[17];
  const float* bnBm = (const float*)d_in[18];
  const float* bnBv = (const float*)d_in[19];
  float* out = (float*)d_out;

  int*   knn_idx = (int*)d_ws;
  float* x1_ws   = (float*)((char*)d_ws + (size_t)8 * 4096 * 3 * sizeof(int));

  dim3 g1(32, 8);
  dgcnn_knn_kernel<<<g1, 256, 0, stream>>>(xg, knn_idx);

  dim3 g2(8, 3);
  dgcnn_conv_max_kernel<<<g2, 256, 0, stream>>>(xg, knn_idx, w1,
                                                bn1g, bn1b, bn1m, bn1v, x1_ws);

  dgcnn_tail_kernel<<<8, 256, 0, stream>>>(x1_ws, wA, bA, wB, bB, wC, bC,
                                           bnAg, bnAb, bnAm, bnAv,
                                           bnBg, bnBb, bnBm, bnBv, out);
}